// PILoss_82669530514041
// MI455X (gfx1250) — compile-verified
//
#include <hip/hip_runtime.h>
#include <stdint.h>

#define Bb 16
#define Tt 32
#define Hh 256
#define Ww 256
#define TILE 32
#define TILES_H (Hh / TILE)              // 8
#define NBLK (Bb * Tt * TILES_H)         // 4096
#define BLK_PER_BATCH (Tt * TILES_H)     // 256

// Low 32 bits of a flat pointer into LDS = group-segment byte offset
// (flat shared addr = {SHARED_BASE_HI, as3_offset}).
__device__ __forceinline__ uint32_t lds_off_of(const void* p) {
    return (uint32_t)(uintptr_t)p;
}

// CDNA5 async global->LDS copy, 16 bytes per lane. Tracked with ASYNCcnt.
__device__ __forceinline__ void async_b128(uint32_t ldsoff, const float* g) {
    asm volatile("global_load_async_to_lds_b128 %0, %1, off"
                 :: "v"(ldsoff), "v"(g) : "memory");
}

__device__ __forceinline__ void wait_async0() {
    asm volatile("s_wait_asynccnt 0" ::: "memory");
}

// torch.gradient edge_order=2: 3-point taps + coefficients, result *= 1/(2h)
__device__ __forceinline__ void stencil3(int i, int n, int& a, int& b, int& c,
                                         float& ca, float& cb, float& cc) {
    if (i == 0)          { a = 0;     b = 1;     c = 2;     ca = -3.f; cb =  4.f; cc = -1.f; }
    else if (i == n - 1) { a = n - 3; b = n - 2; c = n - 1; ca =  1.f; cb = -4.f; cc =  3.f; }
    else                 { a = i - 1; b = i;     c = i + 1; ca = -1.f; cb =  0.f; cc =  1.f; }
}

__global__ __launch_bounds__(256)
void piloss_main(const float* __restrict__ pg, const float* __restrict__ Kg,
                 float* __restrict__ part_num, float* __restrict__ part_den) {
    __shared__ __align__(16) float sp[TILE + 4][Ww];  // p rows r0-2 .. r0+TILE+1 (clamped)
    __shared__ __align__(16) float sk[TILE + 2][Ww];  // K rows r0-1 .. r0+TILE   (clamped)
    __shared__ float rnum[256];
    __shared__ float rden[256];

    const int tid  = threadIdx.x;
    const int blk  = blockIdx.x;
    const int tile = blk & (TILES_H - 1);
    const int t    = (blk >> 3) & (Tt - 1);
    const int b    = blk >> 8;
    const int r0   = tile * TILE;

    const size_t imgBase = ((size_t)b * Tt + t) * (size_t)(Hh * Ww);
    const float* pimg = pg + imgBase;
    const float* kimg = Kg + imgBase;

    // ---- async stage p/K tiles into LDS (b128 chunks, coalesced, 16B aligned) ----
    {
        const int CH_P = (TILE + 4) * (Ww / 4);   // 36*64 = 2304
        for (int q = tid; q < CH_P; q += 256) {
            int row = q >> 6;
            int c4  = (q & 63) << 2;
            int gr  = r0 - 2 + row;
            gr = gr < 0 ? 0 : (gr > Hh - 1 ? Hh - 1 : gr);
            async_b128(lds_off_of(&sp[row][c4]), pimg + (size_t)gr * Ww + c4);
        }
        const int CH_K = (TILE + 2) * (Ww / 4);   // 34*64 = 2176
        for (int q = tid; q < CH_K; q += 256) {
            int row = q >> 6;
            int c4  = (q & 63) << 2;
            int gr  = r0 - 1 + row;
            gr = gr < 0 ? 0 : (gr > Hh - 1 ? Hh - 1 : gr);
            async_b128(lds_off_of(&sk[row][c4]), kimg + (size_t)gr * Ww + c4);
        }
        wait_async0();
    }
    __syncthreads();

    const int w = tid;
    const float inv2hx = 1.0f / (2.0f * 25.0f);   // 0.02
    const float inv2hy = 1.0f / (2.0f * 50.0f);   // 0.01

    float lnum = 0.f, lden = 0.f;

    // column (W) taps are fixed per thread
    int ya, yb, yc; float cya, cyb, cyc;
    stencil3(w, Ww, ya, yb, yc, cya, cyb, cyc);

    // time taps are block-uniform
    int tta, ttb, ttc; float cta, ctb, ctc;
    stencil3(t, Tt, tta, ttb, ttc, cta, ctb, ctc);
    const float* pta = pg + ((size_t)b * Tt + tta) * (size_t)(Hh * Ww);
    const float* ptb = pg + ((size_t)b * Tt + ttb) * (size_t)(Hh * Ww);
    const float* ptc = pg + ((size_t)b * Tt + ttc) * (size_t)(Hh * Ww);
    const bool need_mid = (ctb != 0.f);

    auto P  = [&](int r, int c) -> float { return sp[r - r0 + 2][c]; };
    auto KK = [&](int r, int c) -> float { return sk[r - r0 + 1][c]; };

    auto pdx_at = [&](int r, int c) -> float {   // edge-aware d/dx of p at (r,c)
        int a, bb, cc2; float ca, cb, cc;
        stencil3(r, Hh, a, bb, cc2, ca, cb, cc);
        return (ca * P(a, c) + cb * P(bb, c) + cc * P(cc2, c)) * inv2hx;
    };
    auto pdy_at = [&](int r, int c) -> float {   // edge-aware d/dy of p at (r,c)
        int a, bb, cc2; float ca, cb, cc;
        stencil3(c, Ww, a, bb, cc2, ca, cb, cc);
        return (ca * P(r, a) + cb * P(r, bb) + cc * P(r, cc2)) * inv2hy;
    };

    for (int j = 0; j < TILE; ++j) {
        const int h = r0 + j;

        // prefetch next row's time-neighbor lines (global_prefetch_b8)
        if (j + 1 < TILE) {
            size_t noff = (size_t)(h + 1) * Ww + w;
            __builtin_prefetch(pta + noff, 0, 3);
            __builtin_prefetch(ptc + noff, 0, 3);
            if (need_mid) __builtin_prefetch(ptb + noff, 0, 3);
        }

        int xa, xb, xc; float cxa, cxb, cxc;
        stencil3(h, Hh, xa, xb, xc, cxa, cxb, cxc);

        // ---- x direction ----
        float pdxa = pdx_at(xa, w), pdxb = pdx_at(xb, w), pdxc = pdx_at(xc, w);
        float Kxa = KK(xa, w), Kxb = KK(xb, w), Kxc = KK(xc, w);
        float divk_x = (cxa * Kxa * pdxa + cxb * Kxb * pdxb + cxc * Kxc * pdxc) * inv2hx;
        float p_dxx  = (cxa * pdxa + cxb * pdxb + cxc * pdxc) * inv2hx;
        float K_dx   = (cxa * Kxa + cxb * Kxb + cxc * Kxc) * inv2hx;
        float p_dx   = (h == 0) ? pdxa : ((h == Hh - 1) ? pdxc : pdxb);

        // ---- y direction ----
        float pdya = pdy_at(h, ya), pdyb = pdy_at(h, yb), pdyc = pdy_at(h, yc);
        float Kya = KK(h, ya), Kyb = KK(h, yb), Kyc = KK(h, yc);
        float divk_y = (cya * Kya * pdya + cyb * Kyb * pdyb + cyc * Kyc * pdyc) * inv2hy;
        float p_dyy  = (cya * pdya + cyb * pdyb + cyc * pdyc) * inv2hy;
        float K_dy   = (cya * Kya + cyb * Kyb + cyc * Kyc) * inv2hy;
        float p_dy   = (w == 0) ? pdya : ((w == Ww - 1) ? pdyc : pdyb);

        float Kc    = KK(h, w);
        float divk  = divk_x + divk_y;
        float divk2 = (p_dxx + p_dyy) * Kc + p_dx * K_dx + p_dy * K_dy;
        float m     = (fabsf(divk - divk2) < 100.0f) ? 1.f : 0.f;

        // ---- time gradient (global, coalesced, L2-resident) ----
        size_t off = (size_t)h * Ww + w;
        float pva = pta[off];
        float pvc = ptc[off];
        float pvb = need_mid ? ptb[off] : 0.f;
        float p_dt = (cta * pva + ctb * pvb + ctc * pvc) * 0.5f;

        float diff = divk - p_dt * 2e-4f;   // q = 0 at [32,32] -> no-op
        lnum += m * diff * diff;
        lden += m;
    }

    // ---- deterministic block reduction -> per-block partials ----
    rnum[tid] = lnum; rden[tid] = lden;
    __syncthreads();
    for (int s = 128; s > 0; s >>= 1) {
        if (tid < s) { rnum[tid] += rnum[tid + s]; rden[tid] += rden[tid + s]; }
        __syncthreads();
    }
    if (tid == 0) { part_num[blk] = rnum[0]; part_den[blk] = rden[0]; }
}

__global__ void piloss_finish(const float* __restrict__ part_num,
                              const float* __restrict__ part_den,
                              float* __restrict__ out) {
    __shared__ float acc[Bb];
    int b = threadIdx.x;
    if (b < Bb) {
        float n = 0.f, d = 0.f;
        for (int i = 0; i < BLK_PER_BATCH; ++i) {   // fixed order: deterministic
            n += part_num[b * BLK_PER_BATCH + i];
            d += part_den[b * BLK_PER_BATCH + i];
        }
        acc[b] = n / d;
    }
    __syncthreads();
    if (threadIdx.x == 0) {
        float s = 0.f;
        for (int i = 0; i < Bb; ++i) s += acc[i];
        out[0] = s * (1.0f / Bb);
    }
}

extern "C" void kernel_launch(void* const* d_in, const int* in_sizes, int n_in,
                              void* d_out, int out_size, void* d_ws, size_t ws_size,
                              hipStream_t stream) {
    const float* p = (const float*)d_in[0];
    const float* K = (const float*)d_in[1];
    float* part_num = (float*)d_ws;          // NBLK floats
    float* part_den = part_num + NBLK;       // NBLK floats (32 KB of d_ws total)

    piloss_main<<<NBLK, 256, 0, stream>>>(p, K, part_num, part_den);
    piloss_finish<<<1, 32, 0, stream>>>(part_num, part_den, (float*)d_out);
}